// OutersumModel_7069516169782
// MI455X (gfx1250) — compile-verified
//
#include <hip/hip_runtime.h>

typedef __attribute__((ext_vector_type(2))) float v2f;
typedef __attribute__((ext_vector_type(8))) float v8f;

#define DD      32      // D
#define CRES    128     // C_RES
#define LL      9       // L
#define NB      4       // batches per block
#define ROWS    (NB*LL) // 36 valid rows
#define PROWS   48      // padded to 3 tiles of 16
#define XPITCH  33      // LDS pitch for X (bank-conflict pad)

// ---------------------------------------------------------------------------
// Phase 0: fold weights.
//   Ai[d][c]  = sum_e Wi[d][e] * W1[e][c]
//   Aj[d][c]  = sum_e Wj[d][e] * W1[e][c]
//   Tau[d][c] = sum_e (Wr[d+8][e] + br[e] + bi[e] + bj[e]) * W1[e][c] + b1[c]
//               (Toeplitz relpos term: one_hot(i-j+16) selects row i-j+16 of Wr,
//                with d = i-j+8 in [0,16])
// ws layout (floats): [0,4096) Ai ; [4096,8192) Aj ; [8192,10368) Tau
// ---------------------------------------------------------------------------
__global__ __launch_bounds__(256) void outersum_precompute(
    const float* __restrict__ Wi, const float* __restrict__ bi,
    const float* __restrict__ Wj, const float* __restrict__ bj,
    const float* __restrict__ Wr, const float* __restrict__ br,
    const float* __restrict__ W1, const float* __restrict__ b1,
    float* __restrict__ ws) {
  float* Ai  = ws;
  float* Aj  = ws + DD * CRES;
  float* Tau = ws + 2 * DD * CRES;
  const int tid = threadIdx.x;

  for (int idx = tid; idx < DD * CRES; idx += 256) {
    const int d = idx >> 7, c = idx & 127;
    float si = 0.f, sj = 0.f;
#pragma unroll
    for (int e = 0; e < DD; ++e) {
      const float w1 = W1[e * CRES + c];
      si = fmaf(Wi[d * DD + e], w1, si);
      sj = fmaf(Wj[d * DD + e], w1, sj);
    }
    Ai[idx] = si;
    Aj[idx] = sj;
  }

  for (int idx = tid; idx < 17 * CRES; idx += 256) {
    const int d = idx >> 7, c = idx & 127;
    float s = b1[c];
#pragma unroll
    for (int e = 0; e < DD; ++e) {
      const float coef = Wr[(d + 8) * DD + e] + br[e] + bi[e] + bj[e];
      s = fmaf(coef, W1[e * CRES + c], s);
    }
    Tau[idx] = s;
  }
}

// ---------------------------------------------------------------------------
// Phase 1: per block of NB=4 batches:
//   U = X @ Ai, V = X @ Aj  via v_wmma_f32_16x16x4_f32 (f32 precision kept)
//   H[b][c] = sum_{i,j} relu(U[b*9+i][c] + V[b*9+j][c] + Tau[i-j+8][c])
//   out[b][o] = sum_c H[b][c] * W2[c][o] + 81*b2[o]
// ---------------------------------------------------------------------------
__global__ __launch_bounds__(128) void outersum_main(
    const float* __restrict__ seq, const float* __restrict__ ws,
    const float* __restrict__ W2, const float* __restrict__ b2,
    float* __restrict__ out) {
  __shared__ float Xs[PROWS * XPITCH];   // 48 x 32 (padded pitch 33)
  __shared__ float Us[PROWS * CRES];     // 48 x 128
  __shared__ float Vs[PROWS * CRES];     // 48 x 128
  __shared__ float sacc[NB * 2];

  const float* __restrict__ Ai  = ws;
  const float* __restrict__ Aj  = ws + DD * CRES;
  const float* __restrict__ Tau = ws + 2 * DD * CRES;

  const int tid  = threadIdx.x;
  const int bb   = blockIdx.x * NB;          // first batch of this block
  const int lane = tid & 31;
  const int wave = tid >> 5;
  const int half = lane >> 4;                // 0: lanes 0-15, 1: lanes 16-31
  const int lrow = lane & 15;

  // ---- stage X rows (36 valid, 12 zero-pad) into LDS ----
  const float* xbase = seq + (size_t)bb * LL * DD;
  for (int idx = tid; idx < PROWS * DD; idx += 128) {
    const int r = idx >> 5, col = idx & 31;
    Xs[r * XPITCH + col] = (r < ROWS) ? xbase[r * DD + col] : 0.f;
  }
  __syncthreads();

  // ---- WMMA GEMMs: 48 tiles total (U: tiles 0..23, V: tiles 24..47) ----
  for (int t = wave; t < 48; t += 4) {
    const int mat = t / 24;                  // 0 -> U/Ai, 1 -> V/Aj
    const int rem = t % 24;
    const int rt  = rem >> 3;                // row tile 0..2
    const int ct  = rem & 7;                 // col tile 0..7
    const float* __restrict__ Bmat = mat ? Aj : Ai;

    v8f acc = {0.f, 0.f, 0.f, 0.f, 0.f, 0.f, 0.f, 0.f};
#pragma unroll
    for (int kk = 0; kk < 8; ++kk) {
      // A fragment: 16x4 f32. lanes 0-15 hold K = 4kk+{0,1}, lanes 16-31 K = +{2,3}
      const int k0 = kk * 4 + 2 * half;
      v2f a, bf;
      a[0]  = Xs[(rt * 16 + lrow) * XPITCH + k0];
      a[1]  = Xs[(rt * 16 + lrow) * XPITCH + k0 + 1];
      // B fragment: 4x16 f32, row-striped across lanes (N = lane%16)
      bf[0] = Bmat[(k0)     * CRES + ct * 16 + lrow];
      bf[1] = Bmat[(k0 + 1) * CRES + ct * 16 + lrow];
      acc = __builtin_amdgcn_wmma_f32_16x16x4_f32(
          /*neg_a=*/false, a, /*neg_b=*/false, bf,
          /*c_mod=*/(short)0, acc, /*reuse_a=*/false, /*reuse_b=*/false);
    }
    // C/D layout: VGPR g -> row (g + 8*half), col lane%16
    float* dst = mat ? Vs : Us;
#pragma unroll
    for (int g = 0; g < 8; ++g)
      dst[(rt * 16 + g + 8 * half) * CRES + ct * 16 + lrow] = acc[g];
  }

  if (tid < NB * 2) sacc[tid] = 0.f;
  __syncthreads();

  // ---- elementwise relu outer-sum + channel reduction ----
  const int c = tid;                         // one channel per thread (128)
  float tau[17];
#pragma unroll
  for (int d = 0; d < 17; ++d) tau[d] = Tau[d * CRES + c];
  const float w20 = W2[c * 2 + 0];
  const float w21 = W2[c * 2 + 1];

  for (int b = 0; b < NB; ++b) {
    float u[LL], v[LL];
#pragma unroll
    for (int i = 0; i < LL; ++i) {
      u[i] = Us[(b * LL + i) * CRES + c];
      v[i] = Vs[(b * LL + i) * CRES + c];
    }
    float H = 0.f;
#pragma unroll
    for (int i = 0; i < LL; ++i) {
#pragma unroll
      for (int j = 0; j < LL; ++j) {
        const float z = u[i] + v[j] + tau[i - j + 8];
        H += fmaxf(z, 0.f);
      }
    }
    atomicAdd(&sacc[b * 2 + 0], H * w20);
    atomicAdd(&sacc[b * 2 + 1], H * w21);
  }
  __syncthreads();

  if (tid < NB * 2) {
    const int b = tid >> 1, o = tid & 1;
    out[(size_t)(bb + b) * 2 + o] = sacc[tid] + 81.f * b2[o];
  }
}

// ---------------------------------------------------------------------------
extern "C" void kernel_launch(void* const* d_in, const int* in_sizes, int n_in,
                              void* d_out, int out_size, void* d_ws, size_t ws_size,
                              hipStream_t stream) {
  (void)in_sizes; (void)n_in; (void)out_size; (void)ws_size;
  const float* seq = (const float*)d_in[0];
  // d_in[1] = relpos (unused: analytic Toeplitz structure)
  const float* Wi = (const float*)d_in[2];
  const float* bi = (const float*)d_in[3];
  const float* Wj = (const float*)d_in[4];
  const float* bj = (const float*)d_in[5];
  const float* Wr = (const float*)d_in[6];
  const float* br = (const float*)d_in[7];
  const float* W1 = (const float*)d_in[8];
  const float* b1 = (const float*)d_in[9];
  const float* W2 = (const float*)d_in[10];
  const float* b2 = (const float*)d_in[11];
  float* ws  = (float*)d_ws;
  float* out = (float*)d_out;

  outersum_precompute<<<1, 256, 0, stream>>>(Wi, bi, Wj, bj, Wr, br, W1, b1, ws);

  const int B = 16384;
  outersum_main<<<B / NB, 128, 0, stream>>>(seq, ws, W2, b2, out);
}